// GraphFeatureProcessor_90108413870156
// MI455X (gfx1250) — compile-verified
//
#include <hip/hip_runtime.h>
#include <hip/hip_bf16.h>

// Problem constants (from reference)
#define ALPHA_SL 0.2f
#define BETA_BL  0.5f
#define BS    256
#define HDIM  32
#define WDIM  32
#define BAND  200
#define FEAT  64
#define HWP   (HDIM * WDIM)     // 1024 pixels per batch image
#define MROWS (BS * HWP)        // 262144 GEMM rows
#define KSTEPS (BAND / 4)       // 50 WMMA k-steps (K=4 per step)
#define KPAIRS (BAND / 2)       // 100 packed k-pairs in LDS

typedef __attribute__((ext_vector_type(2))) float v2f;
typedef __attribute__((ext_vector_type(8))) float v8f;

__device__ __forceinline__ float lrelu(float v) { return v > 0.0f ? v : ALPHA_SL * v; }

// ---------------------------------------------------------------------------
// Kernel 1: central-pixel features.
//   c1[b,f]    = lrelu( x[b,16,16,:] . Wk[:,f] )
//   c2[b,f]    = lrelu( x[b,16,16,:] . Wv[:,f] )
//   cfeat[b,f] = lrelu( sum_g c2[b,g] * Wn[f,g] )
// One block per batch image, 64 threads (one per feature).
// ---------------------------------------------------------------------------
__global__ void central_kernel(const float* __restrict__ x,
                               const float* __restrict__ Wk,
                               const float* __restrict__ Wv,
                               const float* __restrict__ Wn,
                               float* __restrict__ c1,
                               float* __restrict__ cfeat) {
  __shared__ float sx[BAND];
  __shared__ float sc2[FEAT];
  const int b = blockIdx.x;
  const int f = threadIdx.x;  // 0..63
  const float* xrow = x + (size_t)(b * HWP + (HDIM / 2) * WDIM + (WDIM / 2)) * BAND;
  for (int k = f; k < BAND; k += FEAT) sx[k] = xrow[k];
  __syncthreads();
  float a1 = 0.0f, a2 = 0.0f;
  for (int k = 0; k < BAND; ++k) {
    const float xv = sx[k];
    a1 = fmaf(xv, Wk[k * FEAT + f], a1);
    a2 = fmaf(xv, Wv[k * FEAT + f], a2);
  }
  a1 = lrelu(a1);
  a2 = lrelu(a2);
  c1[b * FEAT + f] = a1;
  sc2[f] = a2;
  __syncthreads();
  float cf = 0.0f;
  for (int g = 0; g < FEAT; ++g) cf = fmaf(sc2[g], Wn[f * FEAT + g], cf);
  cfeat[b * FEAT + f] = lrelu(cf);
}

// ---------------------------------------------------------------------------
// Kernel 2: dual GEMM with WMMA f32 16x16x4 + fused mean-corr dot + h2 store.
// 256 threads = 8 waves; each wave owns one 16-row tile of M and all 8
// 16x16 N-tiles (4 for the Wk projection, 4 for the Wv projection).
//
// LDS holds the weights PRE-SWIZZLED as k-pairs:
//   ldsK2[k2*64 + n] = { Wk[2*k2][n], Wk[2*k2+1][n] }   (float2, 8B aligned)
// so every B fragment (lane n+16*half needs K = kb+2*half, kb+2*half+1 at
// column n) is a single ds_load_b64 into an even-aligned VGPR pair — no
// repacking movs, half the DS ops of the 2ADDR_B32 scheme.
//
// WMMA f32 16x16x4 VGPR layouts (ISA 7.12.2):
//   A (16x4):  lane n (+16*half) holds M=n, K = 2*half + vgpr   -> float2 load
//   B (4x16):  lane n (+16*half) holds N=n, K = 2*half + vgpr
//   C (16x16): vgpr v on lane n (+16*half) holds M = v + 8*half, N = n
//
// x is streamed once -> non-temporal loads (keep 192MB L2 free for h2,
// which kernel 4 re-reads).
// ---------------------------------------------------------------------------
__global__ void __launch_bounds__(256) gemm_attn_kernel(
    const float* __restrict__ x,
    const float* __restrict__ Wk,
    const float* __restrict__ Wv,
    const float* __restrict__ c1,
    float* __restrict__ mc,   // [MROWS] mean correlation
    float* __restrict__ h2) { // [MROWS][FEAT] lrelu(x @ Wv)
  extern __shared__ v2f lds2[];  // ldsK2[KPAIRS*FEAT] ++ ldsV2[KPAIRS*FEAT]
  v2f* ldsK2 = lds2;
  v2f* ldsV2 = lds2 + KPAIRS * FEAT;
  const int tid = threadIdx.x;

  // Cooperative staging with k-pair swizzle. Consecutive threads hit
  // consecutive columns n -> two coalesced 256B streams per k2 row.
  for (int j = tid; j < KPAIRS * FEAT; j += 256) {
    const int k2 = j >> 6;   // 0..99
    const int n  = j & 63;
    v2f pk = {Wk[(2 * k2) * FEAT + n], Wk[(2 * k2 + 1) * FEAT + n]};
    ldsK2[j] = pk;
    v2f pv = {Wv[(2 * k2) * FEAT + n], Wv[(2 * k2 + 1) * FEAT + n]};
    ldsV2[j] = pv;
  }
  __syncthreads();

  const int wave = tid >> 5;
  const int lane = tid & 31;
  const int n    = lane & 15;
  const int half = lane >> 4;
  const int tile = blockIdx.x * 8 + wave;   // 16384 tiles total
  const int m0   = tile * 16;
  const int b    = m0 >> 10;                // 1024 rows per batch; tiles never straddle b

  // central1 values at this lane's column positions (one per N-tile)
  const float c1v0 = c1[b * FEAT +  0 + n];
  const float c1v1 = c1[b * FEAT + 16 + n];
  const float c1v2 = c1[b * FEAT + 32 + n];
  const float c1v3 = c1[b * FEAT + 48 + n];

  v8f accK[4], accV[4];
#pragma unroll
  for (int t = 0; t < 4; ++t) {
    accK[t] = (v8f)0.0f;
    accV[t] = (v8f)0.0f;
  }

  // A fragment source: row (m0+n), k pair starting at 2*half (8B aligned).
  const float* arow = x + (size_t)(m0 + n) * BAND + 2 * half;

  for (int ks = 0; ks < KSTEPS; ++ks) {
    const v2f a = __builtin_nontemporal_load((const v2f*)(arow + ks * 4));
    const v2f* bkp = ldsK2 + (ks * 2 + half) * FEAT + n;
    const v2f* bvp = ldsV2 + (ks * 2 + half) * FEAT + n;
#pragma unroll
    for (int t = 0; t < 4; ++t) {
      const v2f fk = bkp[t * 16];   // single ds_load_b64, even VGPR pair
      accK[t] = __builtin_amdgcn_wmma_f32_16x16x4_f32(
          false, a, false, fk, (short)0, accK[t], false, false);
      const v2f fv = bvp[t * 16];
      accV[t] = __builtin_amdgcn_wmma_f32_16x16x4_f32(
          false, a, false, fv, (short)0, accV[t], false, false);
    }
  }

  // Leaky ReLU on both projections.
#pragma unroll
  for (int t = 0; t < 4; ++t) {
#pragma unroll
    for (int v = 0; v < 8; ++v) {
      accK[t][v] = lrelu(accK[t][v]);
      accV[t][v] = lrelu(accV[t][v]);
    }
  }

  // mean_corr[row] = (1/64) * h1[row,:] . central1[b,:]
  // Row M = v + 8*half lives on one 16-lane group; butterfly-reduce over N.
#pragma unroll
  for (int v = 0; v < 8; ++v) {
    float p = accK[0][v] * c1v0 + accK[1][v] * c1v1 +
              accK[2][v] * c1v2 + accK[3][v] * c1v3;
    p += __shfl_xor(p, 1, 16);
    p += __shfl_xor(p, 2, 16);
    p += __shfl_xor(p, 4, 16);
    p += __shfl_xor(p, 8, 16);
    if (n == 0) mc[m0 + half * 8 + v] = p * (1.0f / FEAT);
  }

  // Store h2 tile (default RT policy -> stays L2-resident for kernel 4).
  // Each (t,v) is a 64B-contiguous run per 16-lane group.
#pragma unroll
  for (int t = 0; t < 4; ++t) {
#pragma unroll
    for (int v = 0; v < 8; ++v) {
      const int row = m0 + half * 8 + v;
      h2[(size_t)row * FEAT + t * 16 + n] = accV[t][v];
    }
  }
}

// ---------------------------------------------------------------------------
// Kernel 3: softmax over the BATCH axis (256) per pixel (1024 pixels).
// One block per pixel, 256 threads (one per batch element).
// ---------------------------------------------------------------------------
__global__ void softmax_b_kernel(const float* __restrict__ mc,
                                 float* __restrict__ attn) {
  __shared__ float red[BS];
  const int hw = blockIdx.x;
  const int b  = threadIdx.x;
  const float v = mc[b * HWP + hw];
  red[b] = v;
  __syncthreads();
  for (int s = 128; s > 0; s >>= 1) {
    if (b < s) red[b] = fmaxf(red[b], red[b + s]);
    __syncthreads();
  }
  const float mx = red[0];
  __syncthreads();
  const float e = expf(v - mx);
  red[b] = e;
  __syncthreads();
  for (int s = 128; s > 0; s >>= 1) {
    if (b < s) red[b] += red[b + s];
    __syncthreads();
  }
  attn[b * HWP + hw] = e / red[0];
}

// ---------------------------------------------------------------------------
// Kernel 4: neighbour_feat[b,f] = sum_hw h2[b,hw,f] * attn[b,hw];
//           out = BETA*central_feat + (1-BETA)*neighbour_feat.
// One block per batch image; 256 threads = 4 hw-slices x 64 features.
// h2 is dead after this read -> non-temporal loads.
// ---------------------------------------------------------------------------
__global__ void neighbour_kernel(const float* __restrict__ h2,
                                 const float* __restrict__ attn,
                                 const float* __restrict__ cfeat,
                                 float* __restrict__ out) {
  __shared__ float red[256];
  const int b   = blockIdx.x;
  const int tid = threadIdx.x;
  const int f   = tid & 63;
  const int c   = tid >> 6;  // 0..3
  float acc = 0.0f;
  for (int hw = c; hw < HWP; hw += 4) {
    const float a = attn[b * HWP + hw];
    const float hval =
        __builtin_nontemporal_load(h2 + ((size_t)b * HWP + hw) * FEAT + f);
    acc = fmaf(hval, a, acc);
  }
  red[tid] = acc;
  __syncthreads();
  if (c == 0) {
    const float s = red[f] + red[64 + f] + red[128 + f] + red[192 + f];
    out[b * FEAT + f] =
        BETA_BL * cfeat[b * FEAT + f] + (1.0f - BETA_BL) * s;
  }
}

// ---------------------------------------------------------------------------
// Host-side launch. Workspace layout (floats):
//   [0,16384)                      central1 (lrelu'd)
//   [16384,32768)                  central_feat (lrelu'd)
//   [32768, 32768+262144)          mean_corr
//   [+262144, +2*262144)           attn
//   [32768+2*262144, +16777216)    h2        -> ~69.3 MB total
// ---------------------------------------------------------------------------
extern "C" void kernel_launch(void* const* d_in, const int* in_sizes, int n_in,
                              void* d_out, int out_size, void* d_ws,
                              size_t ws_size, hipStream_t stream) {
  const float* x  = (const float*)d_in[0];
  const float* Wk = (const float*)d_in[1];
  const float* Wv = (const float*)d_in[2];
  const float* Wn = (const float*)d_in[3];
  float* out = (float*)d_out;

  float* ws    = (float*)d_ws;
  float* c1    = ws;
  float* cfeat = ws + 16384;
  float* mc    = ws + 32768;
  float* attn  = ws + 32768 + MROWS;
  float* h2    = ws + 32768 + 2 * (size_t)MROWS;

  central_kernel<<<BS, FEAT, 0, stream>>>(x, Wk, Wv, Wn, c1, cfeat);

  const size_t lds_bytes = 2 * KPAIRS * FEAT * sizeof(v2f);  // 102400 B
  gemm_attn_kernel<<<MROWS / 16 / 8, 256, lds_bytes, stream>>>(x, Wk, Wv, c1,
                                                               mc, h2);

  softmax_b_kernel<<<HWP, BS, 0, stream>>>(mc, attn);

  neighbour_kernel<<<BS, 256, 0, stream>>>(h2, attn, cfeat, out);
}